// IMoE_42021960024095
// MI455X (gfx1250) — compile-verified
//
#include <hip/hip_runtime.h>
#include <hip/hip_bf16.h>

// MoE forward for MI455X (gfx1250, wave32).
// Math: only experts 0/1 ever have nonzero scale (top_p quirk), so
//   final = [s0*x | s1*x] @ [W0; W1]^T   (M=16384, N=1024, K=2048)
//   out   = final @ out_W^T              (M=16384, N=1024, K=1024)
// Both GEMMs run on V_WMMA_F32_16X16X4_F32. HBM-bound: ~140MB of true HBM
// traffic (final[N,H]=64MB lives in the 192MB L2) -> ~6us floor at 23.3TB/s,
// vs ~2us of f32 WMMA compute. Tiles are staged with the CDNA5 async
// global->LDS path (GLOBAL_LOAD_ASYNC_TO_LDS_B128, ASYNCcnt) and DOUBLE
// BUFFERED: tile t+1's copies are in flight while tile t computes, using
// s_wait_asynccnt 0x8 (async loads complete in order, ISA ch10 4.1).

typedef __attribute__((ext_vector_type(2))) float v2f;
typedef __attribute__((ext_vector_type(4))) float v4f;
typedef __attribute__((ext_vector_type(8))) float v8f;

#define TOP_P 0.3f
#define BM 128
#define BN 128
#define BK 32

// ---------------- Pass 0: gate softmax + multiplicity scales ----------------
// 8 lanes per token (one per expert), 32 tokens per 256-thread block.
__global__ __launch_bounds__(256)
void gate_kernel(const float* __restrict__ x, const float* __restrict__ gW,
                 float* __restrict__ scale, int N, int D) {
  const int t   = threadIdx.x;
  const int tok = blockIdx.x * 32 + (t >> 3);
  const int e   = t & 7;
  if (tok >= N) return;
  const float* xr = x  + (size_t)tok * D;
  const float* wr = gW + (size_t)e   * D;
  float acc = 0.f;
#pragma unroll 4
  for (int d = 0; d < D; d += 4) {
    v4f xv = *(const v4f*)(xr + d);
    v4f wv = *(const v4f*)(wr + d);
    acc = fmaf(xv.x, wv.x, acc);
    acc = fmaf(xv.y, wv.y, acc);
    acc = fmaf(xv.z, wv.z, acc);
    acc = fmaf(xv.w, wv.w, acc);
  }
  // softmax over the 8 lanes of this token (sub-wave groups of 8)
  float m = acc;
  m = fmaxf(m, __shfl_xor(m, 1, 8));
  m = fmaxf(m, __shfl_xor(m, 2, 8));
  m = fmaxf(m, __shfl_xor(m, 4, 8));
  float ex = __expf(acc - m);
  float s = ex;
  s += __shfl_xor(s, 1, 8);
  s += __shfl_xor(s, 2, 8);
  s += __shfl_xor(s, 4, 8);
  const float p = ex / s;
  int c1 = (p > TOP_P) ? 1 : 0;
  c1 += __shfl_xor(c1, 1, 8);
  c1 += __shfl_xor(c1, 2, 8);
  c1 += __shfl_xor(c1, 4, 8);
  if (e == 0) scale[(size_t)tok * 2 + 0] = (float)(8 - c1) * p;
  if (e == 1) scale[(size_t)tok * 2 + 1] = (float)c1 * p;
}

// Issue one 16-byte async global->LDS copy (per active lane).
__device__ __forceinline__ void async_copy_b128(unsigned lds_addr,
                                                const float* gptr) {
  asm volatile("global_load_async_to_lds_b128 %0, %1, off"
               :: "v"(lds_addr), "v"((unsigned long long)(uintptr_t)gptr)
               : "memory");
}

// ---------------- WMMA GEMM: Dst[M,NC] = A' @ B' ----------------
// mode 0 (expert pass): K=2048 concat; A'(m,k) = A[m, k&1023] * scale[m, k>>10]
//                       B'(k,n) = Bsrc[(k>>10)*H*D + n*1024 + (k&1023)]
// mode 1 (out pass):    K=1024;  A'(m,k) = A[m,k];  B'(k,n) = Bsrc[n*1024 + k]
// 256 threads = 8 waves; block tile 128x128, k-tile 32; each wave owns a
// 64x32 tile = 4x2 WMMA accumulators; 64 v_wmma per LDS tile per wave.
// LDS tiles are K-contiguous so async b128 staging needs no transpose; the
// expert scales are applied to the A fragments (packed v_pk_mul co-executes).
__global__ __launch_bounds__(256)
void moe_gemm(const float* __restrict__ A, const float* __restrict__ Bsrc,
              const float* __restrict__ scale, float* __restrict__ Dst,
              int M, int NC, int K, int mode) {
  __shared__ float As[2][BM][BK + 4];  // [buf][row][k]
  __shared__ float Bs[2][BN][BK + 4];  // [buf][col][k] (K-contig, like memory)
  (void)M;

  const int tid   = threadIdx.x;
  const int bm    = blockIdx.x * BM;
  const int bn    = blockIdx.y * BN;
  const int wave  = tid >> 5;
  const int lane  = tid & 31;
  const int wm    = (wave >> 2) * 64;  // wave M offset: 0 / 64
  const int wn    = (wave & 3) * 32;   // wave N offset: 0..96
  const int lm    = lane & 15;         // row (A) / col (B) within 16
  const int khalf = lane >> 4;         // VGPR pair selects K = 2*khalf + {0,1}

  v8f acc[4][2] = {};                  // 4 (M) x 2 (N) 16x16 f32 tiles

  // per-thread staging coordinates (2 threads per row/col, 16 floats each)
  const int arow = tid >> 1;
  const int ac0  = (tid & 1) * 16;
  const int bcol = tid >> 1;
  const int bk0  = (tid & 1) * 16;
  const unsigned as_dst[2] = {(unsigned)(uintptr_t)&As[0][arow][ac0],
                              (unsigned)(uintptr_t)&As[1][arow][ac0]};
  const unsigned bs_dst[2] = {(unsigned)(uintptr_t)&Bs[0][bcol][bk0],
                              (unsigned)(uintptr_t)&Bs[1][bcol][bk0]};

  // hoist per-row expert scales for this wave's 4 M-subtiles (mode 0 only)
  float sc0[4], sc1[4];
#pragma unroll
  for (int i = 0; i < 4; ++i) {
    if (mode == 0) {
      const int r = bm + wm + i * 16 + lm;
      sc0[i] = scale[(size_t)r * 2 + 0];
      sc1[i] = scale[(size_t)r * 2 + 1];
    } else {
      sc0[i] = 1.0f;
      sc1[i] = 1.0f;
    }
  }

  // stage one k-tile (8 async b128 per thread) into LDS buffer `buf`
  auto stage = [&](int k0, int buf) {
    const int h = k0 >> 10;
    const float* aptr =
        A + (size_t)(bm + arow) * 1024 + (mode == 0 ? (k0 & 1023) : k0);
    const float* bptr = (mode == 0)
        ? Bsrc + (size_t)h * (1024 * 1024) + (size_t)(bn + bcol) * 1024 +
              (k0 & 1023)
        : Bsrc + (size_t)(bn + bcol) * 1024 + k0;
#pragma unroll
    for (int v = 0; v < 4; ++v) {
      async_copy_b128(as_dst[buf] + v * 16, aptr + ac0 + v * 4);
      async_copy_b128(bs_dst[buf] + v * 16, bptr + bk0 + v * 4);
    }
    // hint tile k0+BK toward L2 while these copies fly (global_prefetch_b8)
    __builtin_prefetch(aptr + ac0 + BK, 0, 0);
    __builtin_prefetch(bptr + bk0 + BK, 0, 0);
  };

  stage(0, 0);                         // prologue: fill buffer 0

  int t = 0;
  for (int k0 = 0; k0 < K; k0 += BK, ++t) {
    const int cur  = t & 1;
    const int half = k0 >> 10;
    const bool more = (k0 + BK) < K;

    // issue next tile into the other buffer (safe: barrier at end of the
    // previous iteration guarantees no wave still reads it)
    if (more) {
      stage(k0 + BK, cur ^ 1);
      // 16 async ops outstanding; in-order completion => <=8 means the
      // current tile's copies have landed while next tile's stay in flight
      asm volatile("s_wait_asynccnt 0x8" ::: "memory");
    } else {
      asm volatile("s_wait_asynccnt 0x0" ::: "memory");
    }
    __syncthreads();                   // all waves staged current buffer

    // ---- WMMA inner loop on buffer `cur` ----
#pragma unroll
    for (int kk = 0; kk < BK; kk += 4) {
      v2f afrag[4];
#pragma unroll
      for (int i = 0; i < 4; ++i) {
        const int r = wm + i * 16 + lm;              // A: lane = M row
        const float s = half ? sc1[i] : sc0[i];
        afrag[i].x = As[cur][r][kk + 2 * khalf + 0] * s;  // VGPR = K, K+1
        afrag[i].y = As[cur][r][kk + 2 * khalf + 1] * s;
      }
      v2f bfrag[2];
#pragma unroll
      for (int j = 0; j < 2; ++j) {
        const int c = wn + j * 16 + lm;              // B: lane = N col
        bfrag[j].x = Bs[cur][c][kk + 2 * khalf + 0];
        bfrag[j].y = Bs[cur][c][kk + 2 * khalf + 1];
      }
#pragma unroll
      for (int i = 0; i < 4; ++i)
#pragma unroll
        for (int j = 0; j < 2; ++j)
          acc[i][j] = __builtin_amdgcn_wmma_f32_16x16x4_f32(
              false, afrag[i], false, bfrag[j], (short)0, acc[i][j],
              false, false);
    }
    __syncthreads();                   // buffer free for re-staging
  }

  // ---- epilogue: 16x16 f32 C/D layout -> global ----
#pragma unroll
  for (int i = 0; i < 4; ++i) {
#pragma unroll
    for (int j = 0; j < 2; ++j) {
      const int col = bn + wn + j * 16 + lm;
#pragma unroll
      for (int r = 0; r < 8; ++r) {
        const int row = bm + wm + i * 16 + khalf * 8 + r;
        Dst[(size_t)row * NC + col] = acc[i][j][r];
      }
    }
  }
}

extern "C" void kernel_launch(void* const* d_in, const int* in_sizes, int n_in,
                              void* d_out, int out_size, void* d_ws, size_t ws_size,
                              hipStream_t stream) {
  (void)in_sizes; (void)n_in; (void)out_size; (void)ws_size;
  const float* x    = (const float*)d_in[0];  // [4,4096,1024]
  const float* gW   = (const float*)d_in[1];  // [8,1024]
  const float* expW = (const float*)d_in[2];  // [8,1024,1024]
  const float* outW = (const float*)d_in[3];  // [1024,1024]
  float* out = (float*)d_out;                 // [4,4096,1024]

  const int N = 4 * 4096, D = 1024, H = 1024;
  float* scale  = (float*)d_ws;               // [N,2]
  float* finalb = scale + (size_t)N * 2;      // [N,H] = 64MB

  gate_kernel<<<N / 32, 256, 0, stream>>>(x, gW, scale, N, D);

  dim3 g1(N / BM, H / BN);   // expert GEMM: K = 2048 (concat experts 0,1)
  moe_gemm<<<g1, 256, 0, stream>>>(x, expW, scale, finalb, N, H, 2048, 0);

  dim3 g2(N / BM, D / BN);   // output GEMM: K = 1024
  moe_gemm<<<g2, 256, 0, stream>>>(finalb, outW, nullptr, out, N, D, 1024, 1);
}